// TINShift_23845658427976
// MI455X (gfx1250) — compile-verified
//
#include <hip/hip_runtime.h>

// Temporal Interlace Shift for MI455X (gfx1250).
// out[n,t,c,:] = x[n, t+shift[n, c/32], c, :] if in range else 0.
// Pure HBM-bandwidth problem (~0.8 GB traffic, ~35us floor at 23.3 TB/s).
// One 256-thread block per contiguous 12.25 KB output row; NT b128 streams;
// unsigned bit-decode of indices; loads clustered for MLP.

constexpr unsigned N_  = 8;
constexpr unsigned T_  = 16;
constexpr unsigned C_  = 256;
constexpr unsigned HW_ = 3136;        // floats per row (divisible by 4)
constexpr unsigned G_  = 8;
constexpr unsigned ROW_V4 = HW_ / 4;  // 784 float4 per row = 3*256 + 16
constexpr unsigned BLK = 256;         // 8 wave32
constexpr unsigned TAIL = ROW_V4 - 3 * BLK;   // 16

typedef float v4f __attribute__((ext_vector_type(4)));

__global__ __launch_bounds__(BLK) void tin_shift_kernel(
    const float* __restrict__ x,
    const int*   __restrict__ shift,
    float*       __restrict__ out) {
  const unsigned row = blockIdx.x;        // flat over [N, T, C]
  const unsigned c   = row & (C_ - 1);    // row % 256
  const unsigned nt  = row >> 8;          // n*T + t
  const unsigned t   = nt & (T_ - 1);     // nt % 16
  const unsigned n   = nt >> 4;           // nt / 16
  const unsigned g   = c >> 5;            // c / 32

  // Block-uniform scalar load of the shift value.
  const int ts = (int)t + shift[(n << 3) | g];

  const unsigned tid = threadIdx.x;
  v4f* __restrict__ dst = (v4f*)out + (size_t)row * ROW_V4;
  const bool tail = tid < TAIL;

  if (ts >= 0 && ts < (int)T_) {
    const v4f* __restrict__ src =
        (const v4f*)x + (size_t)(((n << 4) + (unsigned)ts) * C_ + c) * ROW_V4;
    // Prefetch the far end of the row while the first loads issue
    // (global_prefetch_b8; speculative, counter-free, OOB-safe).
    __builtin_prefetch(&src[tid + 3 * BLK], 0, 1);
    // Cluster loads (3-4 b128 in flight per wave) before any store.
    v4f a = __builtin_nontemporal_load(&src[tid]);
    v4f b = __builtin_nontemporal_load(&src[tid + BLK]);
    v4f d = __builtin_nontemporal_load(&src[tid + 2 * BLK]);
    v4f e;
    if (tail) e = __builtin_nontemporal_load(&src[tid + 3 * BLK]);
    __builtin_nontemporal_store(a, &dst[tid]);
    __builtin_nontemporal_store(b, &dst[tid + BLK]);
    __builtin_nontemporal_store(d, &dst[tid + 2 * BLK]);
    if (tail) __builtin_nontemporal_store(e, &dst[tid + 3 * BLK]);
  } else {
    const v4f z = {0.f, 0.f, 0.f, 0.f};
    __builtin_nontemporal_store(z, &dst[tid]);
    __builtin_nontemporal_store(z, &dst[tid + BLK]);
    __builtin_nontemporal_store(z, &dst[tid + 2 * BLK]);
    if (tail) __builtin_nontemporal_store(z, &dst[tid + 3 * BLK]);
  }
}

extern "C" void kernel_launch(void* const* d_in, const int* in_sizes, int n_in,
                              void* d_out, int out_size, void* d_ws, size_t ws_size,
                              hipStream_t stream) {
  (void)in_sizes; (void)n_in; (void)out_size; (void)d_ws; (void)ws_size;
  const float* x     = (const float*)d_in[0];
  const int*   shift = (const int*)d_in[1];
  float*       out   = (float*)d_out;

  const unsigned rows = N_ * T_ * C_;     // 32768 blocks
  tin_shift_kernel<<<rows, BLK, 0, stream>>>(x, shift, out);
}